// MD_BiDirectional_LSTM_56667798503979
// MI455X (gfx1250) — compile-verified
//
#include <hip/hip_runtime.h>
#include <cmath>

typedef __attribute__((ext_vector_type(2))) float v2f;
typedef __attribute__((ext_vector_type(8))) float v8f;

#define DEV static __device__ __forceinline__

namespace {
constexpr int Bn = 64;          // batch
constexpr int Fn = 128;         // features
constexpr int Hn = 512;         // hidden
constexpr int Ln = 16;          // teacher-forced steps per direction
constexpr int WGS_PER_DIR = 16; // workgroups per direction
constexpr int THREADS = 256;    // 8 wave32 per workgroup
// per-direction scratch (floats): h0,c0,h1,c1 (4 * 64*512) + xbuf (64*128)
constexpr size_t DIR_WS = 4 * (size_t)Bn * Hn + (size_t)Bn * Fn; // 139264
constexpr size_t WS_FLOATS = 2 * DIR_WS + 4;                     // + barrier counters
}

DEV v2f ld2(const float* __restrict__ p) { return *(const v2f* __restrict__)p; }

DEV v8f zero8() {
  v8f z;
#pragma unroll
  for (int i = 0; i < 8; ++i) z[i] = 0.0f;
  return z;
}

DEV v8f wmma4(v2f a, v2f b, v8f c) {
  // D = A(16x4 f32) * B(4x16 f32) + C(16x16 f32), full fp32 matrix pipe
  return __builtin_amdgcn_wmma_f32_16x16x4_f32(false, a, false, b, (short)0, c,
                                               false, false);
}

DEV float sigf(float x) { return 1.0f / (1.0f + expf(-x)); }

// Accumulate A(row block) x W^T into 4 gate accumulators (i,f,g,o blocks of W).
DEV void mma_quad(const float* __restrict__ A, int lda,
                  const float* __restrict__ W, int ldw, int K,
                  int arow, int wrow, int khalf,
                  v8f& g0, v8f& g1, v8f& g2, v8f& g3) {
  const float* ap = A + (size_t)arow * lda + khalf;
  const float* wp = W + (size_t)wrow * ldw + khalf;
  const size_t gs = (size_t)Hn * ldw;
#pragma unroll 4
  for (int k = 0; k < K; k += 4) {
    v2f a = ld2(ap + k);
    g0 = wmma4(a, ld2(wp + k), g0);
    g1 = wmma4(a, ld2(wp + gs + k), g1);
    g2 = wmma4(a, ld2(wp + 2 * gs + k), g2);
    g3 = wmma4(a, ld2(wp + 3 * gs + k), g3);
  }
}

DEV void mma_one(const float* __restrict__ A, int lda,
                 const float* __restrict__ W, int ldw, int K,
                 int arow, int wrow, int khalf, v8f& acc) {
  const float* ap = A + (size_t)arow * lda + khalf;
  const float* wp = W + (size_t)wrow * ldw + khalf;
#pragma unroll 4
  for (int k = 0; k < K; k += 4) {
    acc = wmma4(ld2(ap + k), ld2(wp + k), acc);
  }
}

// Monotonic-epoch grid barrier across the 16 workgroups of one direction.
DEV void dir_sync(unsigned* __restrict__ cnt, unsigned target) {
  __syncthreads();
  if (threadIdx.x == 0) {
    __threadfence(); // release this WG's stores to device scope
    __hip_atomic_fetch_add(cnt, 1u, __ATOMIC_RELEASE, __HIP_MEMORY_SCOPE_AGENT);
    while (__hip_atomic_load(cnt, __ATOMIC_ACQUIRE, __HIP_MEMORY_SCOPE_AGENT) <
           target) {
      __builtin_amdgcn_s_sleep(2);
    }
  }
  __syncthreads();
  __threadfence(); // acquire: drop stale near-cache lines before re-reading
}

__global__ void __launch_bounds__(THREADS)
zero_kernel(float* __restrict__ out, size_t n_out, float* __restrict__ ws,
            size_t n_ws) {
  size_t i = (size_t)blockIdx.x * blockDim.x + threadIdx.x;
  size_t stride = (size_t)gridDim.x * blockDim.x;
  for (size_t j = i; j < n_out; j += stride) out[j] = 0.0f;
  for (size_t j = i; j < n_ws; j += stride) ws[j] = 0.0f;
}

__global__ void __launch_bounds__(THREADS)
lstm_main(const int* __restrict__ tptr, const float* __restrict__ x,
          const float* __restrict__ Wi0, const float* __restrict__ Wh0,
          const float* __restrict__ bi0, const float* __restrict__ bh0,
          const float* __restrict__ Wi1, const float* __restrict__ Wh1,
          const float* __restrict__ bi1, const float* __restrict__ bh1,
          const float* __restrict__ oW, const float* __restrict__ ob,
          float* __restrict__ out, float* __restrict__ ws) {
  const int dir = blockIdx.x >> 4; // 0 = forward, 1 = backward
  const int wg = blockIdx.x & 15;
  const int lane = threadIdx.x & 31;
  const int wave = threadIdx.x >> 5;
  const int gw = wg * 8 + wave; // 0..127 global wave in direction
  const int mt = gw >> 5;       // batch tile 0..3  (rows mt*16..mt*16+15)
  const int nh = gw & 31;       // unit tile 0..31  (cols nh*16..nh*16+15)

  const int total = *tptr + 2 * Ln; // output sequence length (256)
  const float inv = 1.0f / (float)(total - 1);

  float* base = ws + (size_t)dir * DIR_WS;
  float* h0 = base;
  float* c0 = base + (size_t)Bn * Hn;
  float* h1 = base + 2 * (size_t)Bn * Hn;
  float* c1 = base + 3 * (size_t)Bn * Hn;
  float* xbuf = base + 4 * (size_t)Bn * Hn; // current input / x_last (B x F)
  unsigned* cnt = (unsigned*)(ws + 2 * DIR_WS) + dir;

  const int l15 = lane & 15;
  const int khalf = (lane >> 4) * 2; // K sub-offset of this half-wave
  const int rowadd = (lane >> 4) * 8; // C/D row offset of this half-wave
  const int col = nh * 16 + l15;      // this lane's output column (units)

  unsigned epoch = 0;

  // ---- emit pred[0] (pure copy of the first teacher input) ----
  {
    int tg = (dir == 0) ? 0 : (total - 1);
    float wa = (float)(total - 1 - tg) * inv;
    float wb = (float)tg * inv;
    float wgt = ((dir == 0) ? wa : wb) / (wa + wb);
    for (int e = threadIdx.x + wg * THREADS; e < Bn * Fn;
         e += WGS_PER_DIR * THREADS) {
      int b = e >> 7, f = e & 127;
      float flip = (f < Fn / 2) ? 1.0f : -1.0f;
      float raw, val;
      if (dir == 0) {
        raw = x[((size_t)b * 2 * Ln + 0) * Fn + f];
        val = raw;
      } else {
        raw = x[((size_t)b * 2 * Ln + (2 * Ln - 1)) * Fn + f] * flip;
        val = raw * flip; // un-flip on emission
      }
      atomicAdd(&out[((size_t)b * total + tg) * Fn + f], wgt * val);
    }
  }

  const int steps = total - 1; // 255 recurrent steps
  for (int s = 0; s < steps; ++s) {
    // ---- stage teacher-forced input (time-flip + +/- flip for backward) ----
    if (s < Ln) {
      for (int e = threadIdx.x + wg * THREADS; e < Bn * Fn;
           e += WGS_PER_DIR * THREADS) {
        int b = e >> 7, f = e & 127;
        float v;
        if (dir == 0) {
          v = x[((size_t)b * 2 * Ln + s) * Fn + f];
        } else {
          v = x[((size_t)b * 2 * Ln + (2 * Ln - 1 - s)) * Fn + f];
          if (f >= Fn / 2) v = -v;
        }
        xbuf[e] = v;
      }
      dir_sync(cnt, (++epoch) * WGS_PER_DIR);
    }

    // ---- layer 0: gates = xbuf @ Wi0^T + h0 @ Wh0^T + b ; fused cell ----
    {
      v8f gi = zero8(), gf = zero8(), gg = zero8(), go = zero8();
      mma_quad(xbuf, Fn, Wi0, Fn, Fn, mt * 16, nh * 16, khalf, gi, gf, gg, go);
      mma_quad(h0, Hn, Wh0, Hn, Hn, mt * 16, nh * 16, khalf, gi, gf, gg, go);
      float bI = bi0[col] + bh0[col];
      float bF = bi0[Hn + col] + bh0[Hn + col];
      float bG = bi0[2 * Hn + col] + bh0[2 * Hn + col];
      float bO = bi0[3 * Hn + col] + bh0[3 * Hn + col];
#pragma unroll
      for (int r = 0; r < 8; ++r) {
        int row = mt * 16 + r + rowadd;
        int idx = row * Hn + col;
        float iv = gi[r] + bI, fv = gf[r] + bF;
        float gv = gg[r] + bG, ov = go[r] + bO;
        float cc = sigf(fv) * c0[idx] + sigf(iv) * tanhf(gv);
        float hh = sigf(ov) * tanhf(cc);
        c0[idx] = cc;
        h0[idx] = hh;
      }
    }
    dir_sync(cnt, (++epoch) * WGS_PER_DIR);

    // ---- layer 1: gates = h0 @ Wi1^T + h1 @ Wh1^T + b ; fused cell ----
    {
      v8f gi = zero8(), gf = zero8(), gg = zero8(), go = zero8();
      mma_quad(h0, Hn, Wi1, Hn, Hn, mt * 16, nh * 16, khalf, gi, gf, gg, go);
      mma_quad(h1, Hn, Wh1, Hn, Hn, mt * 16, nh * 16, khalf, gi, gf, gg, go);
      float bI = bi1[col] + bh1[col];
      float bF = bi1[Hn + col] + bh1[Hn + col];
      float bG = bi1[2 * Hn + col] + bh1[2 * Hn + col];
      float bO = bi1[3 * Hn + col] + bh1[3 * Hn + col];
#pragma unroll
      for (int r = 0; r < 8; ++r) {
        int row = mt * 16 + r + rowadd;
        int idx = row * Hn + col;
        float iv = gi[r] + bI, fv = gf[r] + bF;
        float gv = gg[r] + bG, ov = go[r] + bO;
        float cc = sigf(fv) * c1[idx] + sigf(iv) * tanhf(gv);
        float hh = sigf(ov) * tanhf(cc);
        c1[idx] = cc;
        h1[idx] = hh;
      }
    }
    dir_sync(cnt, (++epoch) * WGS_PER_DIR);

    // ---- projection: x_next = xbuf + h1 @ oW^T + ob ; blend into out ----
    if (nh < Fn / 16) { // 8 feature tiles -> 32 of 128 waves
      v8f acc = zero8();
      mma_one(h1, Hn, oW, Hn, Hn, mt * 16, nh * 16, khalf, acc);
      int f = nh * 16 + l15;
      float bOut = ob[f];
      float flip = (f < Fn / 2) ? 1.0f : -1.0f;
      int p = s + 1;
      int tg = (dir == 0) ? p : (total - 1 - p);
      float wa = (float)(total - 1 - tg) * inv;
      float wb = (float)tg * inv;
      float wgt = ((dir == 0) ? wa : wb) / (wa + wb);
#pragma unroll
      for (int r = 0; r < 8; ++r) {
        int row = mt * 16 + r + rowadd; // batch index
        float nx = xbuf[row * Fn + f] + acc[r] + bOut;
        xbuf[row * Fn + f] = nx;
        float val = (dir == 0) ? nx : nx * flip;
        atomicAdd(&out[((size_t)row * total + tg) * Fn + f], wgt * val);
      }
    }
    dir_sync(cnt, (++epoch) * WGS_PER_DIR);
  }
}

extern "C" void kernel_launch(void* const* d_in, const int* in_sizes, int n_in,
                              void* d_out, int out_size, void* d_ws,
                              size_t ws_size, hipStream_t stream) {
  const int* tptr = (const int*)d_in[0];
  const float* x = (const float*)d_in[1];
  const float* Wi0 = (const float*)d_in[2];
  const float* Wh0 = (const float*)d_in[3];
  const float* bi0 = (const float*)d_in[4];
  const float* bh0 = (const float*)d_in[5];
  const float* Wi1 = (const float*)d_in[6];
  const float* Wh1 = (const float*)d_in[7];
  const float* bi1 = (const float*)d_in[8];
  const float* bh1 = (const float*)d_in[9];
  const float* oW = (const float*)d_in[10];
  const float* ob = (const float*)d_in[11];
  float* out = (float*)d_out;
  float* ws = (float*)d_ws;

  // Re-initialize output accumulator, LSTM state and barrier counters every
  // launch (graph-replay safe: no state survives between calls).
  zero_kernel<<<512, THREADS, 0, stream>>>(out, (size_t)out_size, ws,
                                           WS_FLOATS);

  // 2 directions x 16 cooperating workgroups; per-direction atomic grid sync.
  lstm_main<<<2 * WGS_PER_DIR, THREADS, 0, stream>>>(
      tptr, x, Wi0, Wh0, bi0, bh0, Wi1, Wh1, bi1, bh1, oW, ob, out, ws);
}